// MyGCN_44839458570483
// MI455X (gfx1250) — compile-verified
//
#include <hip/hip_runtime.h>

#define N_NODES 100000
#define D_HID 128
#define D_OUTF 64
#define N_EDGES 1600000

typedef float v2f __attribute__((ext_vector_type(2)));
typedef float v8f __attribute__((ext_vector_type(8)));

// ---------------------------------------------------------------- utilities

__global__ void fill_zero_f4_kernel(float4* __restrict__ p, long long n4) {
    long long i = (long long)blockIdx.x * blockDim.x + threadIdx.x;
    if (i < n4) p[i] = make_float4(0.f, 0.f, 0.f, 0.f);
}

__global__ void deg_kernel(const long long* __restrict__ col,
                           float* __restrict__ deg, int E) {
    int i = blockIdx.x * blockDim.x + threadIdx.x;
    if (i < E) atomicAdd(&deg[(int)col[i]], 1.0f);
}

__global__ void inv_sqrt_kernel(float* __restrict__ deg, int n) {
    int i = blockIdx.x * blockDim.x + threadIdx.x;
    if (i < n) {
        float d = deg[i];
        deg[i] = (d > 0.f) ? rsqrtf(d) : 0.f;
    }
}

// ------------------------------------------------- WMMA fp32 GEMM (16x16x4)
// Computes Out[N x M]:
//   B_TRANSPOSED=true : Out[n,m] = sum_k A[n,k] * B[m*K + k]   (Linear weight [M,K])
//   B_TRANSPOSED=false: Out[n,m] = sum_k A[n,k] * B[k*M + m]   (plain A @ B)
// One wave per 16x16 output tile; K stepped by 4 with V_WMMA_F32_16X16X4_F32.
//
// Fragment layouts per CDNA5 ISA 7.12.2:
//   A (16x4 f32, 2 VGPR): lane = m + 16*(k>=2), vgpr = k&1
//   B (4x16 f32, 2 VGPR): lane = n + 16*(k>=2), vgpr = k&1
//   C/D (16x16 f32, 8 VGPR): lane = n + 16*(m>=8), vgpr = m&7
template <int K, int M, bool B_TRANSPOSED>
__global__ void gemm_wmma_kernel(const float* __restrict__ A,
                                 const float* __restrict__ B,
                                 const float* __restrict__ bias, // nullptr -> none
                                 float* __restrict__ Out,
                                 int nTiles) {
    int tile = blockIdx.x * (blockDim.x >> 5) + (threadIdx.x >> 5);
    if (tile >= nTiles) return;

    constexpr int tilesM = M / 16;
    const int tm = (tile / tilesM) * 16;   // row base
    const int tn = (tile % tilesM) * 16;   // col base

    const int lane = threadIdx.x & 31;
    const int half = lane >> 4;            // selects K pair {0,1} vs {2,3}
    const int l15  = lane & 15;

    const float* aPtr = A + (size_t)(tm + l15) * K + half * 2;
    v8f c = {};

    if (B_TRANSPOSED) {
        const float* bPtr = B + (size_t)(tn + l15) * K + half * 2;
#pragma unroll
        for (int k0 = 0; k0 < K; k0 += 4) {
            v2f a = *(const v2f*)(aPtr + k0);
            v2f b = *(const v2f*)(bPtr + k0);
            c = __builtin_amdgcn_wmma_f32_16x16x4_f32(
                    false, a, false, b, (short)0, c, false, false);
        }
    } else {
        const float* bPtr = B + (size_t)(half * 2) * M + (tn + l15);
#pragma unroll
        for (int k0 = 0; k0 < K; k0 += 4) {
            v2f a = *(const v2f*)(aPtr + k0);
            v2f b;
            b.x = bPtr[(size_t)k0 * M];
            b.y = bPtr[(size_t)(k0 + 1) * M];
            c = __builtin_amdgcn_wmma_f32_16x16x4_f32(
                    false, a, false, b, (short)0, c, false, false);
        }
    }

    const float bv = bias ? bias[tn + l15] : 0.f;
    float* oPtr = Out + (size_t)(tm + half * 8) * M + tn + l15;
#pragma unroll
    for (int v = 0; v < 8; ++v)
        oPtr[(size_t)v * M] = c[v] + bv;
}

// -------------------------------------------- edge gather * norm -> scatter
// One wave per edge; lane owns a float4 feature chunk (128 = 32 lanes * 4).
__global__ void scatter_kernel(const float* __restrict__ h,
                               const long long* __restrict__ rowIdx,
                               const long long* __restrict__ colIdx,
                               const float* __restrict__ dis,
                               float* __restrict__ agg, int E) {
    int e = blockIdx.x * (blockDim.x >> 5) + (threadIdx.x >> 5);
    if (e >= E) return;
    const int lane = threadIdx.x & 31;

    const int r = (int)rowIdx[e];
    const int c = (int)colIdx[e];
    const float nrm = dis[r] * dis[c];

    const float4 v = *((const float4*)(h + (size_t)r * D_HID) + lane);
    float* ap = agg + (size_t)c * D_HID + lane * 4;
    atomicAdd(ap + 0, nrm * v.x);
    atomicAdd(ap + 1, nrm * v.y);
    atomicAdd(ap + 2, nrm * v.z);
    atomicAdd(ap + 3, nrm * v.w);
}

__global__ void bias_relu_kernel(float* __restrict__ h,
                                 const float* __restrict__ bias,
                                 long long total) {
    long long i = (long long)blockIdx.x * blockDim.x + threadIdx.x;
    if (i < total) {
        float v = h[i] + bias[i & (D_HID - 1)];
        h[i] = v > 0.f ? v : 0.f;
    }
}

// ------------------------------------------------------------------- launch

extern "C" void kernel_launch(void* const* d_in, const int* in_sizes, int n_in,
                              void* d_out, int out_size, void* d_ws, size_t ws_size,
                              hipStream_t stream) {
    (void)in_sizes; (void)n_in; (void)out_size; (void)ws_size;

    const float*     x      = (const float*)d_in[0];
    const long long* eidx   = (const long long*)d_in[1];
    const float*     lin0_w = (const float*)d_in[2];
    const float*     bias0  = (const float*)d_in[3];
    const float*     lin1_w = (const float*)d_in[4];
    const float*     bias1  = (const float*)d_in[5];
    const float*     out_w  = (const float*)d_in[6];
    const float*     out_b  = (const float*)d_in[7];
    float*           out    = (float*)d_out;

    const long long* rowIdx = eidx;             // edge_index[0]
    const long long* colIdx = eidx + N_EDGES;   // edge_index[1]

    // Workspace: dis[N] | hbuf[N*128] | agg[N*128]   (~103 MB)
    char* ws = (char*)d_ws;
    size_t off = 0;
    float* dis = (float*)(ws + off);
    off += ((size_t)N_NODES * sizeof(float) + 511) & ~(size_t)511;
    float* hbuf = (float*)(ws + off);
    off += (size_t)N_NODES * D_HID * sizeof(float);
    float* agg = (float*)(ws + off);

    const long long featN = (long long)N_NODES * D_HID;      // 12.8M
    const int tilesHid = (N_NODES / 16) * (D_HID / 16);      // 50000
    const int tilesOut = (N_NODES / 16) * (D_OUTF / 16);     // 25000

    // ---- normalization coefficients -------------------------------------
    fill_zero_f4_kernel<<<(N_NODES / 4 + 255) / 256, 256, 0, stream>>>(
        (float4*)dis, N_NODES / 4);
    deg_kernel<<<(N_EDGES + 255) / 256, 256, 0, stream>>>(colIdx, dis, N_EDGES);
    inv_sqrt_kernel<<<(N_NODES + 255) / 256, 256, 0, stream>>>(dis, N_NODES);

    // ---- conv layer 0 ----------------------------------------------------
    gemm_wmma_kernel<D_HID, D_HID, true>
        <<<(tilesHid + 7) / 8, 256, 0, stream>>>(x, lin0_w, nullptr, hbuf, tilesHid);
    fill_zero_f4_kernel<<<(int)((featN / 4 + 255) / 256), 256, 0, stream>>>(
        (float4*)agg, featN / 4);
    scatter_kernel<<<(N_EDGES + 7) / 8, 256, 0, stream>>>(
        hbuf, rowIdx, colIdx, dis, agg, N_EDGES);
    bias_relu_kernel<<<(int)((featN + 255) / 256), 256, 0, stream>>>(
        agg, bias0, featN);

    // ---- conv layer 1 (agg holds h1; reuse buffers) ----------------------
    gemm_wmma_kernel<D_HID, D_HID, true>
        <<<(tilesHid + 7) / 8, 256, 0, stream>>>(agg, lin1_w, nullptr, hbuf, tilesHid);
    fill_zero_f4_kernel<<<(int)((featN / 4 + 255) / 256), 256, 0, stream>>>(
        (float4*)agg, featN / 4);
    scatter_kernel<<<(N_EDGES + 7) / 8, 256, 0, stream>>>(
        hbuf, rowIdx, colIdx, dis, agg, N_EDGES);
    bias_relu_kernel<<<(int)((featN + 255) / 256), 256, 0, stream>>>(
        agg, bias1, featN);

    // ---- output head: out = h2 @ out_w + out_b ---------------------------
    gemm_wmma_kernel<D_HID, D_OUTF, false>
        <<<(tilesOut + 7) / 8, 256, 0, stream>>>(agg, out_w, out_b, out, tilesOut);
}